// FreqVMEncoder_8744553415213
// MI455X (gfx1250) — compile-verified
//
#include <hip/hip_runtime.h>
#include <hip/hip_bf16.h>
#include <stdint.h>

// FreqVM encoder for MI455X (gfx1250).
// - Whole cv table (36x8x128 f32 = 147456 B) staged into dynamic LDS with 36
//   back-to-back CDNA5 async global->LDS DMA ops (ASYNCcnt-tracked). The
//   instruction OFFSET field is added to BOTH the LDS and global addresses
//   (CDNA5 ISA), so one base VGPR pair + immediate offsets
//   covers the whole copy with zero per-op address arithmetic.
//   147 KB LDS/workgroup is only possible on CDNA5's 320 KB WGP LDS;
//   2 blocks/WGP (294 KB) stay co-resident.
// - Per-point `points` load issued before s_wait_asynccnt so its latency
//   hides under the DMA.
// - cm bilinear: i0 clamped to RES-2 so each corner pair is one contiguous
//   b64 gather (2 VMEM ops per bilinear, not 4); cm (18.9 MB) is L2-resident
//   (192 MB L2).
// - The 6 j-features per (f,c) are stored contiguously so the backend merges
//   them into wide stores (confirmed by last round's histogram).

#define NFREQ 6
#define CCH 8
#define RESN 128
#define NROW (NFREQ * 6)              // 36
#define CV_ELEMS (NROW * CCH * RESN)  // 36864 floats = 147456 bytes
#define OUT_STRIDE (3 + NROW * CCH)   // 291
#define STAGE_ITERS (CV_ELEMS * 4 / (256 * 16))  // 36

template <int IT>
__device__ __forceinline__ void stage_cv(unsigned lds_addr, uint64_t gaddr) {
  asm volatile("global_load_async_to_lds_b128 %0, %1, off offset:%2"
               :: "v"(lds_addr), "v"(gaddr), "n"(IT * 4096) : "memory");
  if constexpr (IT + 1 < STAGE_ITERS) stage_cv<IT + 1>(lds_addr, gaddr);
}

__device__ __forceinline__ void lin_coord(float g, int& i0, float& w) {
  // pos = (g+1)*0.5*(RES-1); clamp i0 to [0, RES-2] and fold the edge into w
  // (identical result to the reference's clip(i0,0,127)/clip(i0+1,0,127)).
  float pos = (g + 1.0f) * 63.5f;
  int i = (int)floorf(pos);
  i = i < 0 ? 0 : i;
  i = i > (RESN - 2) ? (RESN - 2) : i;
  w = pos - (float)i;
  i0 = i;
}

__global__ void __launch_bounds__(256)
freqvm_kernel(const float* __restrict__ points,
              const float* __restrict__ cv,
              const float* __restrict__ cm,
              float* __restrict__ out, int n)
{
  extern __shared__ float s_cv[];  // CV_ELEMS floats

  // ---- Issue the full cv -> LDS async DMA (16 B per lane per op). ----
  {
    unsigned lds_addr = (unsigned)(uintptr_t)s_cv + (unsigned)threadIdx.x * 16u;
    uint64_t gaddr = (uint64_t)(uintptr_t)cv + (uint64_t)threadIdx.x * 16u;
    stage_cv<0>(lds_addr, gaddr);
  }

  // ---- Overlap: pull this thread's point while the DMA is in flight. ----
  const int gid = blockIdx.x * 256 + (int)threadIdx.x;
  const bool active = gid < n;
  float px = 0.0f, py = 0.0f, pz = 0.0f;
  if (active) {
    px = points[(size_t)gid * 3 + 0];
    py = points[(size_t)gid * 3 + 1];
    pz = points[(size_t)gid * 3 + 2];
  }

  asm volatile("s_wait_asynccnt 0x0" ::: "memory");
  __syncthreads();
  if (!active) return;

  float* orow = out + (size_t)gid * OUT_STRIDE;
  orow[0] = px; orow[1] = py; orow[2] = pz;

  float freq = 1.0f;  // 2^linspace(0,5,6) = 1,2,4,8,16,32
  for (int f = 0; f < NFREQ; ++f, freq *= 2.0f) {
    float sx, cx, sy, cy, sz, cz;
    __sincosf(px * freq, &sx, &cx);
    __sincosf(py * freq, &sy, &cy);
    __sincosf(pz * freq, &sz, &cz);

    float* obase = orow + 3 + f * 6;  // + c*36 + j per feature

    // e = [sx,sy,sz,cx,cy,cz]; XIDX=[1,2,0,4,5,3]; YIDX=[2,0,1,5,3,4]
#define DO_ROW(J, G, GX, GY)                                                  \
    {                                                                         \
      const int row = f * 6 + (J);                                            \
      int iv, ixx, iyy; float wv, wx, wy;                                     \
      lin_coord((G),  iv,  wv);                                               \
      lin_coord((GX), ixx, wx);                                               \
      lin_coord((GY), iyy, wy);                                               \
      const float* cvr = s_cv + row * (CCH * RESN);                           \
      const float* cmr = cm + (size_t)row * (CCH * RESN * RESN);              \
      const int idx = iyy * RESN + ixx;                                       \
      _Pragma("unroll")                                                       \
      for (int c = 0; c < CCH; ++c) {                                         \
        float v0 = cvr[c * RESN + iv];                                        \
        float v1 = cvr[c * RESN + iv + 1];                                    \
        float vec = fmaf(v1 - v0, wv, v0);                                    \
        const float* pm = cmr + (size_t)c * (RESN * RESN) + idx;              \
        float m00 = pm[0];                                                    \
        float m01 = pm[1];                                                    \
        float m10 = pm[RESN];                                                 \
        float m11 = pm[RESN + 1];                                             \
        float mx0 = fmaf(m01 - m00, wx, m00);                                 \
        float mx1 = fmaf(m11 - m10, wx, m10);                                 \
        float mat = fmaf(mx1 - mx0, wy, mx0);                                 \
        obase[c * NROW + (J)] = vec * mat;                                    \
      }                                                                       \
    }

    DO_ROW(0, sx, sy, sz)
    DO_ROW(1, sy, sz, sx)
    DO_ROW(2, sz, sx, sy)
    DO_ROW(3, cx, cy, cz)
    DO_ROW(4, cy, cz, cx)
    DO_ROW(5, cz, cx, cy)
#undef DO_ROW
  }
}

extern "C" void kernel_launch(void* const* d_in, const int* in_sizes, int n_in,
                              void* d_out, int out_size, void* d_ws, size_t ws_size,
                              hipStream_t stream) {
  const float* points = (const float*)d_in[0];
  const float* cv     = (const float*)d_in[1];
  const float* cm     = (const float*)d_in[2];
  float* out = (float*)d_out;
  const int n = in_sizes[0] / 3;

  const size_t shmem = (size_t)CV_ELEMS * sizeof(float);  // 147456 B
  (void)hipFuncSetAttribute(reinterpret_cast<const void*>(freqvm_kernel),
                            hipFuncAttributeMaxDynamicSharedMemorySize,
                            (int)shmem);

  freqvm_kernel<<<(n + 255) / 256, 256, shmem, stream>>>(points, cv, cm, out, n);
}